// Verify_ans_91130616087028
// MI455X (gfx1250) — compile-verified
//
#include <hip/hip_runtime.h>

#define B_ 32
#define S_ 1024
#define H_ 1024

typedef float v2f __attribute__((ext_vector_type(2)));
typedef float v8f __attribute__((ext_vector_type(8)));

// One workgroup = one (batch, 16-row block). 8 waves of 32 (wave32).
// Phase 1: S_blk[16,1024] = R_blk * R^T  (f32 WMMA, K-loop step 4) -> LDS
// Phase 2: diag=0, row max, exp, row sum (in LDS)
// Phase 3: V_blk[16,1024] = P_blk * R    (f32 WMMA, A from LDS)
// Epilogue: logits[i] = <r_i,w1> + <v_i,w2> + <v_i*r_i,w3> + b
__global__ __launch_bounds__(256) void verify_attn_kernel(
    const float* __restrict__ rep, const float* __restrict__ w3w,
    const float* __restrict__ w3b, float* __restrict__ logits)
{
  extern __shared__ float sm[];
  float* lds_S     = sm;                // 16*1024 floats (S, then P)
  float* lds_red   = sm + 16 * S_;      // 256
  float* lds_m     = lds_red + 256;     // 16
  float* lds_l     = lds_m + 16;        // 16
  float* lds_logit = lds_l + 16;        // 16

  const int tid  = threadIdx.x;
  const int wave = tid >> 5;
  const int lane = tid & 31;
  const int hi   = lane >> 4;   // half-wave select (K pair / M+8)
  const int lo   = lane & 15;   // M or N index within tile

  const int blk = blockIdx.x;
  const int b   = blk >> 6;           // 64 row-blocks per batch
  const int i0  = (blk & 63) << 4;    // first row of this block

  const float* Rb = rep + (size_t)b * S_ * H_;

  const v8f vz = {};
  v8f acc[8];
  #pragma unroll
  for (int t = 0; t < 8; ++t) acc[t] = vz;

  // ---------------- Phase 1: S = R_blk * Rb^T ----------------
  const int n_base = wave * 128;                    // this wave's 8 N-tiles
  const float* arow = Rb + (size_t)(i0 + lo) * H_ + 2 * hi;
  for (int k0 = 0; k0 < H_; k0 += 4) {
    v2f a = *(const v2f*)(arow + k0);               // A[lo, k0+2hi .. +1]
    #pragma unroll
    for (int t = 0; t < 8; ++t) {
      const int n0 = n_base + t * 16;
      v2f bf = *(const v2f*)(Rb + (size_t)(n0 + lo) * H_ + k0 + 2 * hi);
      acc[t] = __builtin_amdgcn_wmma_f32_16x16x4_f32(
          false, a, false, bf, (short)0, acc[t], false, false);
    }
  }
  // write S tiles to LDS: acc[t][r] is element (M = r+8*hi, N = n0+lo)
  #pragma unroll
  for (int t = 0; t < 8; ++t) {
    const int n0 = n_base + t * 16;
    #pragma unroll
    for (int r = 0; r < 8; ++r)
      lds_S[(r + 8 * hi) * S_ + n0 + lo] = acc[t][r];
  }
  __syncthreads();

  // ---------------- Phase 2: diag, softmax stats ----------------
  if (tid < 16) lds_S[tid * S_ + i0 + tid] = 0.0f;   // diagonal logit -> 0
  __syncthreads();

  const int row = tid >> 4, sub = tid & 15;          // 16 threads per row
  float lmax = -3.402823466e38f;
  for (int c = sub; c < S_; c += 16) lmax = fmaxf(lmax, lds_S[row * S_ + c]);
  lds_red[tid] = lmax;
  __syncthreads();
  if (sub == 0) {
    float m = lds_red[tid];
    for (int j = 1; j < 16; ++j) m = fmaxf(m, lds_red[tid + j]);
    lds_m[row] = m;
  }
  __syncthreads();
  const float rmax = lds_m[row];
  float lsum = 0.0f;
  for (int c = sub; c < S_; c += 16) {
    float e = __expf(lds_S[row * S_ + c] - rmax);
    lds_S[row * S_ + c] = e;                         // S -> unnormalized P
    lsum += e;
  }
  lds_red[tid] = lsum;
  __syncthreads();
  if (sub == 0) {
    float l = 0.0f;
    for (int j = 0; j < 16; ++j) l += lds_red[tid + j];
    lds_l[row] = l;
  }
  if (tid < 16) lds_logit[tid] = 0.0f;
  __syncthreads();

  // ---------------- Phase 3: V = P * Rb ----------------
  #pragma unroll
  for (int t = 0; t < 8; ++t) acc[t] = vz;
  const int h_base = wave * 128;                     // this wave's 8 H-tiles
  const float* ldsA = lds_S + lo * S_ + 2 * hi;
  for (int k0 = 0; k0 < S_; k0 += 4) {
    v2f a = *(const v2f*)(ldsA + k0);                // P[lo, k0+2hi .. +1]
    #pragma unroll
    for (int t = 0; t < 8; ++t) {
      const int h0 = h_base + t * 16;
      v2f bf;
      bf.x = Rb[(size_t)(k0 + 2 * hi) * H_ + h0 + lo];
      bf.y = Rb[(size_t)(k0 + 2 * hi + 1) * H_ + h0 + lo];
      acc[t] = __builtin_amdgcn_wmma_f32_16x16x4_f32(
          false, a, false, bf, (short)0, acc[t], false, false);
    }
  }

  // ---------------- Epilogue: logit partial sums ----------------
  #pragma unroll
  for (int r = 0; r < 8; ++r) {
    const int m = r + 8 * hi;
    const float invl = 1.0f / lds_l[m];
    float part = 0.0f;
    #pragma unroll
    for (int t = 0; t < 8; ++t) {
      const int h  = h_base + t * 16 + lo;
      const float v   = acc[t][r] * invl;            // normalized V[m,h]
      const float rmh = Rb[(size_t)(i0 + m) * H_ + h];
      part += rmh * w3w[h] + v * w3w[H_ + h] + v * rmh * w3w[2 * H_ + h];
    }
    atomicAdd(&lds_logit[m], part);
  }
  __syncthreads();
  if (tid < 16) logits[(size_t)b * S_ + i0 + tid] = lds_logit[tid] + w3b[0];
}

// log_softmax over the sequence dim, one workgroup per batch.
__global__ __launch_bounds__(256) void logsoftmax_kernel(
    const float* __restrict__ logits, float* __restrict__ out)
{
  __shared__ float red[256];
  const int b = blockIdx.x, tid = threadIdx.x;
  const float* x = logits + (size_t)b * S_;

  float lm = -3.402823466e38f;
  for (int i = tid; i < S_; i += 256) lm = fmaxf(lm, x[i]);
  red[tid] = lm;
  __syncthreads();
  for (int off = 128; off > 0; off >>= 1) {
    if (tid < off) red[tid] = fmaxf(red[tid], red[tid + off]);
    __syncthreads();
  }
  const float m = red[0];
  __syncthreads();

  float ls = 0.0f;
  for (int i = tid; i < S_; i += 256) ls += __expf(x[i] - m);
  red[tid] = ls;
  __syncthreads();
  for (int off = 128; off > 0; off >>= 1) {
    if (tid < off) red[tid] += red[tid + off];
    __syncthreads();
  }
  const float lse = m + __logf(red[0]);

  for (int i = tid; i < S_; i += 256) out[(size_t)b * S_ + i] = x[i] - lse;
}

extern "C" void kernel_launch(void* const* d_in, const int* in_sizes, int n_in,
                              void* d_out, int out_size, void* d_ws, size_t ws_size,
                              hipStream_t stream) {
  (void)in_sizes; (void)n_in; (void)out_size; (void)ws_size;
  const float* rep = (const float*)d_in[0];   // [32,1024,1024] f32
  const float* w3w = (const float*)d_in[1];   // [1,3072] f32
  const float* w3b = (const float*)d_in[2];   // [1] f32
  float* out    = (float*)d_out;              // [32,1024] f32
  float* logits = (float*)d_ws;               // scratch: 32*1024 f32 = 128 KB

  const size_t smem = (size_t)(16 * S_ + 256 + 16 + 16 + 16) * sizeof(float);
  verify_attn_kernel<<<dim3(B_ * (S_ / 16)), dim3(256), smem, stream>>>(
      rep, w3w, w3b, logits);
  logsoftmax_kernel<<<dim3(B_), dim3(256), 0, stream>>>(logits, out);
}